// SimpleObjectDetector_57354993271018
// MI455X (gfx1250) — compile-verified
//
#include <hip/hip_runtime.h>
#include <hip/hip_bf16.h>

typedef float v2f __attribute__((ext_vector_type(2)));
typedef float v8f __attribute__((ext_vector_type(8)));

// ---------------------------------------------------------------------------
// Fused conv1(3x3,3->64) + ReLU + conv2(3x3,64->128) + ReLU + global-mean.
// One workgroup = one batch b x one 16x16 output tile. 224 = 14*16 exactly.
//   conv1 -> LDS halo tile h[64][18][18] (plain VALU, only ~2% of FLOPs)
//   conv2 -> implicit GEMM with V_WMMA_F32_16X16X4_F32 (full f32 precision)
//            M = out-channel (8 waves x 16), N = position (16 tiles of 16),
//            K = ci*9 + tap, 576 total = 144 steps of K=4.
//   feats accumulated with global f32 atomics (zeroed beforehand).
// ---------------------------------------------------------------------------
__global__ __launch_bounds__(256) void fused_conv_feats_kernel(
    const float* __restrict__ xin,   // [32][3][224][224]
    const float* __restrict__ w1,    // [64][3][3][3]
    const float* __restrict__ b1,    // [64]
    const float* __restrict__ w2,    // [128][64][3][3] == row-major [128][576]
    const float* __restrict__ b2,    // [128]
    float* __restrict__ feats)       // [32][128]
{
    __shared__ float x_lds[3 * 20 * 20];     // x halo (20x20 per input ch)
    __shared__ float w1_lds[64 * 27];
    __shared__ float b1_lds[64];
    __shared__ float b2_lds[128];
    __shared__ float h_lds[64 * 18 * 18];    // conv1 output halo tile

    const int tid = threadIdx.x;
    const int b   = blockIdx.z;
    const int ty0 = blockIdx.y * 16;
    const int tx0 = blockIdx.x * 16;

    // ---- stage weights / biases into LDS -------------------------------
    for (int i = tid; i < 64 * 27; i += 256) w1_lds[i] = w1[i];
    if (tid < 64)  b1_lds[tid] = b1[tid];
    if (tid < 128) b2_lds[tid] = b2[tid];

    // ---- load x halo (20x20x3), zero outside image ---------------------
    for (int idx = tid; idx < 1200; idx += 256) {
        int c   = idx / 400;
        int rem = idx - c * 400;
        int yy  = rem / 20;
        int xx  = rem - yy * 20;
        int gy  = ty0 - 2 + yy;
        int gx  = tx0 - 2 + xx;
        float v = 0.0f;
        if (gy >= 0 && gy < 224 && gx >= 0 && gx < 224)
            v = xin[((size_t)b * 3 + c) * 50176 + (size_t)gy * 224 + gx];
        x_lds[idx] = v;
    }
    __syncthreads();

    // ---- conv1 into h halo (18x18x64); zero outside image (SAME pad of h)
    // 64*18*18 = 20736 = 81 * 256
    for (int i = 0; i < 81; ++i) {
        int idx = tid + 256 * i;
        int ci  = idx / 324;
        int rem = idx - ci * 324;
        int y   = rem / 18;
        int xx  = rem - y * 18;
        int gy  = ty0 - 1 + y;
        int gx  = tx0 - 1 + xx;
        float v = 0.0f;
        if (gy >= 0 && gy < 224 && gx >= 0 && gx < 224) {
            float s = b1_lds[ci];
            const float* wp = &w1_lds[ci * 27];
#pragma unroll
            for (int c = 0; c < 3; ++c)
#pragma unroll
                for (int dy = 0; dy < 3; ++dy)
#pragma unroll
                    for (int dx = 0; dx < 3; ++dx)
                        s = fmaf(x_lds[c * 400 + (y + dy) * 20 + (xx + dx)],
                                 wp[c * 9 + dy * 3 + dx], s);
            v = fmaxf(s, 0.0f);
        }
        h_lds[idx] = v;   // idx == ci*324 + y*18 + xx
    }
    __syncthreads();

    // ---- conv2 as implicit GEMM with f32 WMMA --------------------------
    const int l     = tid & 31;        // lane in wave32
    const int mw    = tid >> 5;        // wave id 0..7 -> out-channel tile
    const int px    = l & 15;          // N lane: x within spatial tile
    const int khalf = (l >> 4) << 1;   // lanes 16-31 carry K=2,3 (ISA layout)
    const int m_row = mw * 16 + (l & 15);
    const float* w2row = w2 + (size_t)m_row * 576;

    v8f acc[16] = {};                  // 16 N-tiles (py = 0..15)

    for (int s = 0; s < 144; ++s) {
        // A fragment: lane m_row, K = 4s+khalf .. +1  (8B-aligned float2)
        v2f a = *(const v2f*)(w2row + 4 * s + khalf);

        // B rows for this lane: k0 = 4s+khalf, k1 = k0+1
        int k0  = 4 * s + khalf;
        int ci0 = k0 / 9;  int t0 = k0 - ci0 * 9;
        int dy0 = t0 / 3;  int dx0 = t0 - dy0 * 3;
        int off0 = ci0 * 324 + dy0 * 18 + dx0 + px;
        int k1  = k0 + 1;
        int ci1 = k1 / 9;  int t1 = k1 - ci1 * 9;
        int dy1 = t1 / 3;  int dx1 = t1 - dy1 * 3;
        int off1 = ci1 * 324 + dy1 * 18 + dx1 + px;

#pragma unroll
        for (int nt = 0; nt < 16; ++nt) {   // nt == py
            v2f bf;
            bf[0] = h_lds[off0 + 18 * nt];
            bf[1] = h_lds[off1 + 18 * nt];
            acc[nt] = __builtin_amdgcn_wmma_f32_16x16x4_f32(
                false, a, false, bf, (short)0, acc[nt], false, false);
        }
    }

    // ---- bias + ReLU + mean reduction ----------------------------------
    // C layout: VGPR r, lanes 0-15 -> M=r, lanes 16-31 -> M=r+8.
    const int   hi8 = (l >> 4) * 8;
    const float inv = 1.0f / (224.0f * 224.0f);
#pragma unroll
    for (int r = 0; r < 8; ++r) {
        const int   ch   = mw * 16 + hi8 + r;
        const float bias = b2_lds[ch];
        float p = 0.0f;
#pragma unroll
        for (int nt = 0; nt < 16; ++nt)
            p += fmaxf(acc[nt][r] + bias, 0.0f);
        // reduce across the 16 lanes of this half-wave (wave32)
        p += __shfl_xor(p, 1, 32);
        p += __shfl_xor(p, 2, 32);
        p += __shfl_xor(p, 4, 32);
        p += __shfl_xor(p, 8, 32);
        if ((l & 15) == 0)
            atomicAdd(&feats[b * 128 + ch], p * inv);
    }
}

// ---------------------------------------------------------------------------
// Per-batch stable sort by -score + sequential NMS. One block per batch.
// ---------------------------------------------------------------------------
__global__ __launch_bounds__(128) void nms_kernel(
    const float* __restrict__ boxes,   // [32][100][4]
    const float* __restrict__ scores,  // [32][100]
    float* __restrict__ out)           // feats|boxes|scores|keep concat
{
    __shared__ float sb[100][4];
    __shared__ float ss[100];
    __shared__ int   ord[100];
    __shared__ float kb[100][4];
    __shared__ float ks[100];
    __shared__ int   keep[100];

    const int b   = blockIdx.x;
    const int tid = threadIdx.x;

    if (tid < 100) {
        ss[tid] = scores[b * 100 + tid];
#pragma unroll
        for (int k = 0; k < 4; ++k)
            sb[tid][k] = boxes[(b * 100 + tid) * 4 + k];
    }
    __syncthreads();

    // stable descending rank (matches jnp.argsort(-scores) tie-break)
    if (tid < 100) {
        float si = ss[tid];
        int rank = 0;
        for (int j = 0; j < 100; ++j) {
            float sj = ss[j];
            rank += (sj > si) || (sj == si && j < tid);
        }
        ord[rank] = tid;
    }
    __syncthreads();

    if (tid < 100) {
        int src = ord[tid];
#pragma unroll
        for (int k = 0; k < 4; ++k) kb[tid][k] = sb[src][k];
        ks[tid]  = ss[src];
        keep[tid] = 1;
    }
    __syncthreads();

    for (int i = 0; i < 99; ++i) {
        if (tid > i && tid < 100 && keep[i]) {
            float x1i = kb[i][0], y1i = kb[i][1], x2i = kb[i][2], y2i = kb[i][3];
            float x1j = kb[tid][0], y1j = kb[tid][1], x2j = kb[tid][2], y2j = kb[tid][3];
            float ai = fmaxf(x2i - x1i, 0.0f) * fmaxf(y2i - y1i, 0.0f);
            float aj = fmaxf(x2j - x1j, 0.0f) * fmaxf(y2j - y1j, 0.0f);
            float xx1 = fmaxf(x1i, x1j), yy1 = fmaxf(y1i, y1j);
            float xx2 = fminf(x2i, x2j), yy2 = fminf(y2i, y2j);
            float inter = fmaxf(xx2 - xx1, 0.0f) * fmaxf(yy2 - yy1, 0.0f);
            float iou = inter / (ai + aj - inter + 1e-9f);
            if (iou > 0.5f) keep[tid] = 0;
        }
        __syncthreads();
    }

    if (tid < 100) {
        const int kp = keep[tid];
        float* ob = out + 4096;                    // final_boxes
#pragma unroll
        for (int k = 0; k < 4; ++k)
            ob[(b * 100 + tid) * 4 + k] = kp ? kb[tid][k] : 0.0f;
        out[16896 + b * 100 + tid] = kp ? ks[tid] : 0.0f;   // final_scores
        out[20096 + b * 100 + tid] = kp ? 1.0f : 0.0f;      // keep
    }
}

__global__ void zero_feats_kernel(float* __restrict__ out)
{
    int i = blockIdx.x * 256 + threadIdx.x;
    if (i < 4096) out[i] = 0.0f;
}

extern "C" void kernel_launch(void* const* d_in, const int* in_sizes, int n_in,
                              void* d_out, int out_size, void* d_ws, size_t ws_size,
                              hipStream_t stream) {
    (void)in_sizes; (void)n_in; (void)out_size; (void)d_ws; (void)ws_size;

    const float* x      = (const float*)d_in[0];
    const float* boxes  = (const float*)d_in[1];
    const float* scores = (const float*)d_in[2];
    const float* w1     = (const float*)d_in[3];
    const float* b1     = (const float*)d_in[4];
    const float* w2     = (const float*)d_in[5];
    const float* b2     = (const float*)d_in[6];
    float* out = (float*)d_out;

    // 1) zero the feats region (conv kernel accumulates with atomics)
    zero_feats_kernel<<<16, 256, 0, stream>>>(out);

    // 2) fused conv1+conv2+mean: 14x14 spatial tiles x 32 batches
    fused_conv_feats_kernel<<<dim3(14, 14, 32), 256, 0, stream>>>(
        x, w1, b1, w2, b2, out);

    // 3) sort + NMS + output assembly: one block per batch
    nms_kernel<<<32, 128, 0, stream>>>(boxes, scores, out);
}